// AdaptiveF0Smoother_73160472920663
// MI455X (gfx1250) — compile-verified
//
#include <hip/hip_runtime.h>
#include <cmath>

#ifndef __has_builtin
#define __has_builtin(x) 0
#endif

namespace {
constexpr int Bsz = 512;          // batch rows
constexpr int Tsz = 32768;        // time steps per row
constexpr int CH  = 256;          // chunk length (time steps per thread)
constexpr int NC  = Tsz / CH;     // 128 chunks per row
constexpr int GRP = NC / 32;      // 4 chunk-groups (waves) per row
constexpr int JT  = 64;           // time steps staged per LDS tile
constexpr int NT  = CH / JT;      // 4 tiles per chunk
constexpr int LSTR = JT + 2;      // 66-float LDS row stride: 8B aligned, conflict-free
constexpr int NCHUNKS = Bsz * NC; // 65536 chunks total
}

// ---- CDNA5 async global<->LDS helpers ----
// Builtin signature (probe-verified via clang diagnostic): takes pointers to
// an 8-byte int vector, global (addrspace(1)) first, LDS (addrspace(3)) second,
// then imm offset and imm cpol.

typedef int v2i_as __attribute__((vector_size(8)));
typedef __attribute__((address_space(1))) v2i_as* g_v2i_ptr;
typedef __attribute__((address_space(3))) v2i_as* l_v2i_ptr;

__device__ __forceinline__ void async_g2l_b64(const float* g, float* l) {
#if __has_builtin(__builtin_amdgcn_global_load_async_to_lds_b64)
  __builtin_amdgcn_global_load_async_to_lds_b64(
      (g_v2i_ptr)(v2i_as*)(void*)const_cast<float*>(g),
      (l_v2i_ptr)(v2i_as*)(void*)l, 0, 0);
#else
  *(float2*)l = *(const float2*)g;
#endif
}

__device__ __forceinline__ void async_l2g_b64(float* g, const float* l) {
#if __has_builtin(__builtin_amdgcn_global_store_async_from_lds_b64)
  __builtin_amdgcn_global_store_async_from_lds_b64(
      (g_v2i_ptr)(v2i_as*)(void*)g,
      (l_v2i_ptr)(v2i_as*)(void*)const_cast<float*>(l), 0, 0);
#else
  *(float2*)g = *(const float2*)l;
#endif
}

__device__ __forceinline__ void wait_async0() {
#if __has_builtin(__builtin_amdgcn_s_wait_asynccnt)
  __builtin_amdgcn_s_wait_asynccnt(0);
#elif defined(__AMDGCN__)
  asm volatile("s_wait_asynccnt 0" ::: "memory");
#endif
#if defined(__AMDGCN__)
  asm volatile("" ::: "memory");
#endif
}

__device__ __forceinline__ void wait_ds0() {
#if __has_builtin(__builtin_amdgcn_s_wait_dscnt)
  __builtin_amdgcn_s_wait_dscnt(0);
#elif defined(__AMDGCN__)
  asm volatile("s_wait_dscnt 0" ::: "memory");
#endif
#if defined(__AMDGCN__)
  asm volatile("" ::: "memory");
#endif
}

// ---- Pass 1: zero-state chunk scan -> per-chunk final state (d0,d1) ----

__global__ __launch_bounds__(32) void f0_pass1(
    const float* __restrict__ x, float* __restrict__ d0, float* __restrict__ d1,
    float a1, float a2) {
  __shared__ float sx[32 * LSTR];
  const int lane   = threadIdx.x;
  const int row    = blockIdx.x / GRP;
  const int chunk0 = (blockIdx.x % GRP) << 5;   // *32
  const float* rowx = x + (size_t)row * Tsz;
  float w1 = 0.f, w2 = 0.f;
  for (int t = 0; t < NT; ++t) {
    const int tb = chunk0 * CH + t * JT + lane * 2;
#pragma unroll
    for (int seg = 0; seg < 32; ++seg)  // 32 lanes x 8B = 256B contiguous per op
      async_g2l_b64(rowx + tb + seg * CH, &sx[seg * LSTR + lane * 2]);
    wait_async0();
    __syncthreads();
    const float* mine = &sx[lane * LSTR];
#pragma unroll
    for (int j = 0; j < JT; ++j) {
      const float xv = mine[j];
      float w0 = fmaf(-a1, w1, xv);
      w0 = fmaf(-a2, w2, w0);
      w2 = w1; w1 = w0;
    }
    __syncthreads();
  }
  const int cidx = row * NC + chunk0 + lane;
  d0[cidx] = w1;   // w0[C-1]
  d1[cidx] = w2;   // w0[C-2]
}

// ---- Pass 2: per-row affine propagation of chunk boundary states ----

__global__ __launch_bounds__(256) void f0_pass2(
    const float* __restrict__ d0, const float* __restrict__ d1,
    float* __restrict__ i0, float* __restrict__ i1,
    float p00, float p01, float p10, float p11) {
  const int row = blockIdx.x * blockDim.x + threadIdx.x;
  if (row >= Bsz) return;
  const float* pd0 = d0 + row * NC;
  const float* pd1 = d1 + row * NC;
  float* pi0 = i0 + row * NC;
  float* pi1 = i1 + row * NC;
  float s0 = 0.f, s1 = 0.f;
  for (int k = 0; k < NC; ++k) {
    pi0[k] = s0;             // init state entering chunk k
    pi1[k] = s1;
    const float t0 = fmaf(p00, s0, fmaf(p01, s1, pd0[k]));
    const float t1 = fmaf(p10, s0, fmaf(p11, s1, pd1[k]));
    s0 = t0; s1 = t1;
  }
}

// ---- Pass 3: re-scan with true init state, apply output + confidence gate ----

__global__ __launch_bounds__(32) void f0_pass3(
    const float* __restrict__ x, const float* __restrict__ cf,
    const float* __restrict__ i0, const float* __restrict__ i1,
    float* __restrict__ y,
    float a1, float a2, float b0, float b1, float b2) {
  __shared__ float sx[32 * LSTR];
  __shared__ float sc[32 * LSTR];
  __shared__ float sy[32 * LSTR];
  const int lane   = threadIdx.x;
  const int row    = blockIdx.x / GRP;
  const int chunk0 = (blockIdx.x % GRP) << 5;
  const size_t rb = (size_t)row * Tsz;
  const float* rowx = x + rb;
  const float* rowc = cf + rb;
  float* rowy = y + rb;
  const int cidx = row * NC + chunk0 + lane;
  float w1 = i0[cidx], w2 = i1[cidx];
  for (int t = 0; t < NT; ++t) {
    const int tb = chunk0 * CH + t * JT + lane * 2;
#pragma unroll
    for (int seg = 0; seg < 32; ++seg) {
      async_g2l_b64(rowx + tb + seg * CH, &sx[seg * LSTR + lane * 2]);
      async_g2l_b64(rowc + tb + seg * CH, &sc[seg * LSTR + lane * 2]);
    }
    wait_async0();           // loads of this tile + stores of previous tile done
    __syncthreads();
    const float* mx = &sx[lane * LSTR];
    const float* mc = &sc[lane * LSTR];
    float* my = &sy[lane * LSTR];
#pragma unroll
    for (int j = 0; j < JT; ++j) {
      const float xv = mx[j];
      float w0 = fmaf(-a1, w1, xv);
      w0 = fmaf(-a2, w2, w0);
      float yv = fmaf(b0, w0, fmaf(b1, w1, b2 * w2));
      yv = (mc[j] > 0.8f) ? xv : yv;
      my[j] = yv;
      w2 = w1; w1 = w0;
    }
    wait_ds0();              // DS writes land before async store engine reads
    __syncthreads();
#pragma unroll
    for (int seg = 0; seg < 32; ++seg)
      async_l2g_b64(rowy + tb + seg * CH, &sy[seg * LSTR + lane * 2]);
  }
  // trailing async stores drained by S_ENDPGM's implicit wait-idle
}

extern "C" void kernel_launch(void* const* d_in, const int* in_sizes, int n_in,
                              void* d_out, int out_size, void* d_ws, size_t ws_size,
                              hipStream_t stream) {
  const float* x  = (const float*)d_in[0];   // f0_frames (B,T) f32
  const float* cf = (const float*)d_in[1];   // confidence (B,T) f32
  float* y  = (float*)d_out;
  float* ws = (float*)d_ws;                  // needs 4*NCHUNKS floats = 1 MB
  float* d0 = ws;
  float* d1 = ws + NCHUNKS;
  float* i0 = ws + 2 * NCHUNKS;
  float* i1 = ws + 3 * NCHUNKS;

  // Biquad coefficients, double math then f32 cast (matches numpy reference)
  const double PI = 3.14159265358979323846;
  const double frame_rate = 44100.0 / 512.0;
  double w0 = 2.0 * PI * 20.0 / frame_rate;
  const double wmax = PI * 0.99;
  if (w0 > wmax) w0 = wmax;
  const double sw = sin(w0), cw = cos(w0);
  const double alpha = sw / (2.0 * 0.707);
  const double a0d = 1.0 + alpha;
  const float b0 = (float)(((1.0 - cw) / 2.0) / a0d);
  const float b1 = (float)((1.0 - cw) / a0d);
  const float b2 = (float)(((1.0 - cw) / 2.0) / a0d);
  const float a1 = (float)((-2.0 * cw) / a0d);
  const float a2 = (float)((1.0 - alpha) / a0d);

  // P = M^CH, M = [[-a1,-a2],[1,0]] : homogeneous propagation over one chunk.
  // Pole radius ~0.42 -> P underflows to ~0 for CH=256; computed generically.
  double m00 = -(double)a1, m01 = -(double)a2, m10 = 1.0, m11 = 0.0;
  double p00 = 1.0, p01 = 0.0, p10 = 0.0, p11 = 1.0;
  for (int i = 0; i < CH; ++i) {
    const double n00 = m00 * p00 + m01 * p10;
    const double n01 = m00 * p01 + m01 * p11;
    const double n10 = m10 * p00 + m11 * p10;
    const double n11 = m10 * p01 + m11 * p11;
    p00 = n00; p01 = n01; p10 = n10; p11 = n11;
  }

  f0_pass1<<<NCHUNKS / 32, 32, 0, stream>>>(x, d0, d1, a1, a2);
  f0_pass2<<<(Bsz + 255) / 256, 256, 0, stream>>>(
      d0, d1, i0, i1, (float)p00, (float)p01, (float)p10, (float)p11);
  f0_pass3<<<NCHUNKS / 32, 32, 0, stream>>>(x, cf, i0, i1, y, a1, a2, b0, b1, b2);
}